// HVAE_86380382257803
// MI455X (gfx1250) — compile-verified
//
#include <hip/hip_runtime.h>

// log(2*pi), log(0.25)
#define LOG2PI_D 1.8378770664093453
#define QLV_F   -1.3862943611198906f

typedef float v2f __attribute__((ext_vector_type(2)));
typedef float v8f __attribute__((ext_vector_type(8)));

// ---------------------------------------------------------------------------
// Exact wave32 sum via V_WMMA_F32_16X16X4_F32 with B = ones.
// A layout (ISA 7.12.2, 32-bit A 16x4): lane L<16 holds A[L][0],A[L][1];
// lane L>=16 holds A[L-16][2],A[L-16][3]. Each lane deposits its partial in
// its first A slot (a[1]=0), so sum(A) == sum of all 32 partials. With B all
// ones, D[m][n] = rowsum_m for every n; per-lane s = sum(d[0..7]) gives
// rowsums 0..7 (low half) / 8..15 (high half); total = s + shfl_xor(s,16).
// Exact f32 (x*1.0 exact, RNE adds). Call sites have EXEC all ones.
// ---------------------------------------------------------------------------
__device__ __forceinline__ float wave_sum_wmma(float p) {
  v2f a; a[0] = p;    a[1] = 0.0f;
  v2f b; b[0] = 1.0f; b[1] = 1.0f;
  v8f c = {};
  v8f d = __builtin_amdgcn_wmma_f32_16x16x4_f32(
      /*neg_a=*/false, a, /*neg_b=*/false, b,
      /*c_mod=*/(short)0, c, /*reuse_a=*/false, /*reuse_b=*/false);
  float s = d[0] + d[1] + d[2] + d[3] + d[4] + d[5] + d[6] + d[7];
  s += __shfl_xor(s, 16, 32);
  return s;  // every lane holds the full 32-lane total
}

// Block-level reduction: WMMA wave sums -> LDS -> thread 0 writes the block
// partial. Deterministic (fixed order). All threads must reach this.
__device__ __forceinline__ void block_store_partial(float p, float* __restrict__ out) {
  __shared__ float wsum[32];
  float s = wave_sum_wmma(p);
  const int lane = threadIdx.x & 31;
  const int wave = threadIdx.x >> 5;
  if (lane == 0) wsum[wave] = s;
  __syncthreads();
  if (threadIdx.x == 0) {
    float t = 0.0f;
    const int nw = blockDim.x >> 5;
    for (int w = 0; w < nw; ++w) t += wsum[w];
    out[blockIdx.x] = t;
  }
}

// BCE-with-logits per-element term: max(x,0) - x*adj + log1p(exp(-|x|))
__device__ __forceinline__ float bce_term(float v, float a) {
  return fmaxf(v, 0.0f) - v * a + log1pf(expf(-fabsf(v)));
}

// ---------------------------------------------------------------------------
// Kernel 1: zero the adjacency bitmask (b128 stores)
// ---------------------------------------------------------------------------
__global__ void zero_u128(uint4* __restrict__ p, long n4) {
  long i = blockIdx.x * (long)blockDim.x + threadIdx.x;
  const long stride = (long)gridDim.x * blockDim.x;
  const uint4 z = make_uint4(0u, 0u, 0u, 0u);
  for (; i < n4; i += stride) p[i] = z;
}

// ---------------------------------------------------------------------------
// Kernel 2: symmetric edge scatter into bitmask. atomicOr is idempotent, so
// duplicate edges / (i,j)&(j,i) pairs / self-loops behave exactly like .set().
// ---------------------------------------------------------------------------
__global__ void scatter_edges(const int* __restrict__ ei, int E, int N,
                              unsigned* __restrict__ mask) {
  const int e = blockIdx.x * blockDim.x + threadIdx.x;
  if (e >= E) return;
  const unsigned i = (unsigned)ei[e];
  const unsigned j = (unsigned)ei[e + E];
  const unsigned p1 = i * (unsigned)N + j;
  const unsigned p2 = j * (unsigned)N + i;
  atomicOr(&mask[p1 >> 5], 1u << (p1 & 31u));
  atomicOr(&mask[p2 >> 5], 1u << (p2 & 31u));
}

// ---------------------------------------------------------------------------
// Kernel 3: BCE-with-logits partial sums over N*N elements, 4 per thread/iter
// via b128 loads. nv = N*N/4. One mask u32 covers the 4 bits (8 lanes share a
// word -> same-address loads coalesce/broadcast).
// ---------------------------------------------------------------------------
__global__ void bce_partial(const float4* __restrict__ x4,
                            const unsigned* __restrict__ mask,
                            long nv, float* __restrict__ partials) {
  long i = blockIdx.x * (long)blockDim.x + threadIdx.x;
  const long stride = (long)gridDim.x * blockDim.x;
  float acc = 0.0f;
  for (; i < nv; i += stride) {
    const float4 v = x4[i];
    const long e = i << 2;                      // element index of v.x
    const unsigned w  = mask[e >> 5];
    const unsigned sh = (unsigned)e & 31u;      // multiple of 4, <= 28
    acc += bce_term(v.x, (float)((w >> (sh + 0u)) & 1u));
    acc += bce_term(v.y, (float)((w >> (sh + 1u)) & 1u));
    acc += bce_term(v.z, (float)((w >> (sh + 2u)) & 1u));
    acc += bce_term(v.w, (float)((w >> (sh + 3u)) & 1u));
  }
  block_store_partial(acc, partials);
}

// ---------------------------------------------------------------------------
// Kernel 4: fused kl_structure + kl_semantic partial sums over N*D elements,
// 4 per thread/iter (nv = N*D/4). D is a multiple of 4, so the 4 consecutive
// d-indices never wrap a row.
// kld = 2*((mu - q)^2 + exp(lv)) - 0.5*(1 + lv - QLV)   [q_logvar = log 0.25]
// ---------------------------------------------------------------------------
__global__ void kl_partial(const float4* __restrict__ zm_n, const float4* __restrict__ zlv_n,
                           const float4* __restrict__ zm_e, const float4* __restrict__ zlv_e,
                           const float* __restrict__ muA,  const float* __restrict__ muB,
                           int D, long nv, float* __restrict__ partials) {
  extern __shared__ float sm[];      // [0,D): mu_Alpha, [D,2D): mu_Beta
  for (int t = threadIdx.x; t < D; t += blockDim.x) {
    sm[t]     = muA[t];
    sm[t + D] = muB[t];
  }
  __syncthreads();
  long i = blockIdx.x * (long)blockDim.x + threadIdx.x;
  const long stride = (long)gridDim.x * blockDim.x;
  float acc = 0.0f;
  for (; i < nv; i += stride) {
    const int d = (int)((i << 2) % D);
    const float4 me = zm_e[i], le = zlv_e[i];
    const float4 mn = zm_n[i], ln = zlv_n[i];
    {
      const float de = me.x - sm[d + 0], dn = mn.x - sm[D + d + 0];
      acc += 2.0f * (de * de + expf(le.x)) - 0.5f * (1.0f + le.x - QLV_F)
           + 2.0f * (dn * dn + expf(ln.x)) - 0.5f * (1.0f + ln.x - QLV_F);
    }
    {
      const float de = me.y - sm[d + 1], dn = mn.y - sm[D + d + 1];
      acc += 2.0f * (de * de + expf(le.y)) - 0.5f * (1.0f + le.y - QLV_F)
           + 2.0f * (dn * dn + expf(ln.y)) - 0.5f * (1.0f + ln.y - QLV_F);
    }
    {
      const float de = me.z - sm[d + 2], dn = mn.z - sm[D + d + 2];
      acc += 2.0f * (de * de + expf(le.z)) - 0.5f * (1.0f + le.z - QLV_F)
           + 2.0f * (dn * dn + expf(ln.z)) - 0.5f * (1.0f + ln.z - QLV_F);
    }
    {
      const float de = me.w - sm[d + 3], dn = mn.w - sm[D + d + 3];
      acc += 2.0f * (de * de + expf(le.w)) - 0.5f * (1.0f + le.w - QLV_F)
           + 2.0f * (dn * dn + expf(ln.w)) - 0.5f * (1.0f + ln.w - QLV_F);
    }
  }
  block_store_partial(acc, partials);
}

// ---------------------------------------------------------------------------
// Kernel 5: final deterministic combine in f64. Single block.
// ---------------------------------------------------------------------------
__global__ void finalize(const float* __restrict__ p1, int n1,
                         const float* __restrict__ p2, int n2,
                         const float* __restrict__ muA, const float* __restrict__ AlphaMu,
                         const float* __restrict__ muB, const float* __restrict__ BetaMu,
                         int D, double invNN, double invND, double invD,
                         float* __restrict__ out) {
  __shared__ double sh[256];
  double s = 0.0;
  for (int i = threadIdx.x; i < n1; i += blockDim.x) s += (double)p1[i];
  sh[threadIdx.x] = s;
  __syncthreads();
  double bce = 0.0;
  if (threadIdx.x == 0)
    for (int t = 0; t < (int)blockDim.x; ++t) bce += sh[t];
  __syncthreads();
  s = 0.0;
  for (int i = threadIdx.x; i < n2; i += blockDim.x) s += (double)p2[i];
  sh[threadIdx.x] = s;
  __syncthreads();
  if (threadIdx.x == 0) {
    double kl = 0.0;
    for (int t = 0; t < (int)blockDim.x; ++t) kl += sh[t];
    // tiny D-length terms:
    //   log_pmu = -0.5*(log(2pi) + mu^2); extra_kl collapses to 2*(mu - q)^2
    double small = 0.0;
    for (int d = 0; d < D; ++d) {
      const double a = (double)muA[d], am = (double)AlphaMu[d];
      const double b = (double)muB[d], bm = (double)BetaMu[d];
      small += -0.5 * (LOG2PI_D + a * a) + 2.0 * (a - am) * (a - am)
             + -0.5 * (LOG2PI_D + b * b) + 2.0 * (b - bm) * (b - bm);
    }
    out[0] = (float)(small * invD + bce * invNN + kl * invND);
  }
}

// ---------------------------------------------------------------------------
// Host side
// ---------------------------------------------------------------------------
extern "C" void kernel_launch(void* const* d_in, const int* in_sizes, int n_in,
                              void* d_out, int out_size, void* d_ws, size_t ws_size,
                              hipStream_t stream) {
  const float* z_mu_n      = (const float*)d_in[0];
  const float* z_logvar_n  = (const float*)d_in[1];
  const float* z_mu_e      = (const float*)d_in[2];
  const float* z_logvar_e  = (const float*)d_in[3];
  const float* Alpha_mu    = (const float*)d_in[4];
  const float* Beta_mu     = (const float*)d_in[5];
  const float* edge_logits = (const float*)d_in[6];
  const float* mu_Alpha    = (const float*)d_in[7];
  const float* mu_Beta     = (const float*)d_in[8];
  const int*   edge_index  = (const int*)d_in[9];   // JAX default x64-off -> int32

  const int  D  = in_sizes[4];                      // 64
  const long ND = (long)in_sizes[0];                // N*D
  const int  N  = (int)(ND / D);                    // 8192
  const long NN = (long)N * (long)N;
  const int  E  = in_sizes[9] / 2;

  // workspace layout: [bitmask: NN/32 u32][partials1: NB1 f32][partials2: NB2 f32]
  const long maskWords = (NN + 31) / 32;            // 2,097,152 (8 MB)
  const int  NB1 = 4096;
  const int  NB2 = 512;
  unsigned* mask      = (unsigned*)d_ws;
  float*    partials1 = (float*)((char*)d_ws + maskWords * sizeof(unsigned));
  float*    partials2 = partials1 + NB1;

  zero_u128<<<1024, 256, 0, stream>>>((uint4*)mask, maskWords / 4);
  scatter_edges<<<(E + 255) / 256, 256, 0, stream>>>(edge_index, E, N, mask);
  bce_partial<<<NB1, 256, 0, stream>>>((const float4*)edge_logits, mask, NN / 4, partials1);
  kl_partial<<<NB2, 256, 2 * D * (int)sizeof(float), stream>>>(
      (const float4*)z_mu_n, (const float4*)z_logvar_n,
      (const float4*)z_mu_e, (const float4*)z_logvar_e,
      mu_Alpha, mu_Beta, D, ND / 4, partials2);
  finalize<<<1, 256, 0, stream>>>(partials1, NB1, partials2, NB2,
                                  mu_Alpha, Alpha_mu, mu_Beta, Beta_mu, D,
                                  1.0 / (double)NN, 1.0 / (double)ND, 1.0 / (double)D,
                                  (float*)d_out);
}